// SelfAttentiveWordExtractor_53523882443268
// MI455X (gfx1250) — compile-verified
//
#include <hip/hip_runtime.h>
#include <math.h>

typedef __attribute__((ext_vector_type(2))) float v2f;
typedef __attribute__((ext_vector_type(8))) float v8f;
typedef __attribute__((ext_vector_type(4))) unsigned int u32x4;
typedef __attribute__((ext_vector_type(8))) int i32x8;
typedef __attribute__((ext_vector_type(4))) int i32x4;

#define Bc 8
#define Tc 2048
#define Dc 1024
#define Sc 512
#define Wc 16
#define ROWSTRIDE 1040   // 1024 data + 4x4 pad floats (one 4-dword pad per 256-dword segment)

#if __has_builtin(__builtin_amdgcn_tensor_load_to_lds)
#define USE_TDM 1
#else
#define USE_TDM 0
#endif

// Logical column d -> physical LDS column within a row (TDM pads 4 dwords
// after every 256-dword segment; 16-wide tiles never straddle a segment).
__device__ __forceinline__ int colOff(int d) { return d + ((d >> 8) << 2); }

// Dynamic smem layout (floats):
//   sRows : 16*ROWSTRIDE  (gathered span rows, TDM-padded layout)
//   sW    : 1024          (att_w copy)
//   sPart : 128           (logit partial sums)
//   sLog  : 16            (per-row logits)
//   sProb : 16            (softmax weights, masks folded in)

__global__ __launch_bounds__(128)
void span_attend_kernel(const float* __restrict__ text,
                        const int* __restrict__ word_indices,
                        const int* __restrict__ word_mask,
                        const int* __restrict__ word_indices_mask,
                        const float* __restrict__ att_w,
                        const float* __restrict__ att_b,
                        float* __restrict__ out)
{
  extern __shared__ float smem[];
  float* sRows = smem;                        // 16*ROWSTRIDE
  float* sW    = smem + 16 * ROWSTRIDE;       // 1024
  float* sPart = sW + 1024;                   // 128
  float* sLog  = sPart + 128;                 // 16
  float* sProb = sLog + 16;                   // 16

  const int span = blockIdx.x;                // 0 .. B*S-1
  const int b    = span / Sc;
  const int tid  = threadIdx.x;
  const int lane = tid & 31;
  const long ibase = (long)span * Wc;

  // ---- att_w -> LDS (256 float4, 128 threads x 2) ----
  {
    const float4* wsrc = (const float4*)att_w;
    float4* wdst = (float4*)sW;
    wdst[tid]       = wsrc[tid];
    wdst[tid + 128] = wsrc[tid + 128];
  }

  const int w = tid >> 3;      // row 0..15 (logit phase)
  const int p = tid & 7;       // sub-part 0..7

#if USE_TDM
  // ---- TDM gather: one tensor_load_to_lds pulls all 16 rows into LDS ----
  if (tid < 32) {
    // Pack 16 clamped row indices as 16-bit values (uniform -> scalar loads).
    unsigned pk[8];
    #pragma unroll
    for (int j = 0; j < 8; ++j) {
      int i0 = word_indices[ibase + 2 * j];     if (i0 < 0) i0 = 0;
      int i1 = word_indices[ibase + 2 * j + 1]; if (i1 < 0) i1 = 0;
      unsigned v = (unsigned)(i0 & 0xFFFF) | ((unsigned)(i1 & 0xFFFF) << 16);
      pk[j] = (unsigned)__builtin_amdgcn_readfirstlane((int)v);
    }
    unsigned long long ga =
        (unsigned long long)(const void*)(text + (long)b * Tc * Dc);
    unsigned ga_lo = (unsigned)__builtin_amdgcn_readfirstlane((int)(ga & 0xFFFFFFFFu));
    unsigned ga_hi = (unsigned)__builtin_amdgcn_readfirstlane((int)((ga >> 32) & 0x01FFFFFFu));
    unsigned lds0  = __builtin_amdgcn_groupstaticsize();   // dynamic LDS starts here (0)

    u32x4 g0;
    g0[0] = 0x80000001u;          // count=1 | gather_mode=1 | 16-bit indices
    g0[1] = lds0;                 // lds_addr (bytes), sRows at dyn-LDS base
    g0[2] = ga_lo;                // global_addr[31:0]
    g0[3] = ga_hi | 0x80000000u;  // global_addr[56:32] | type=2 ("image")

    i32x8 g1;
    g1[0] = 0x07D20000;  // data_size=4B | pad_enable | pad_interval=256dw | pad_amount=4dw
    g1[1] = 0x04000000;  // tensor_dim0 = 1024 (bits [31:16] = low half)
    g1[2] = 0x08000000;  // tensor_dim1 = 2048 (bits [31:16] = low half)
    g1[3] = 0x04000000;  // tile_dim0   = 1024 (bits [31:16])
    g1[4] = 16;          // tile_dim1   = #valid gather indices
    g1[5] = 1024;        // tensor_dim0_stride[31:0]
    g1[6] = 0;
    g1[7] = 0;

    i32x4 g2, g3;
    g2[0] = (int)pk[0]; g2[1] = (int)pk[1]; g2[2] = (int)pk[2]; g2[3] = (int)pk[3];
    g3[0] = (int)pk[4]; g3[1] = (int)pk[5]; g3[2] = (int)pk[6]; g3[3] = (int)pk[7];

#if defined(__clang_major__) && (__clang_major__ >= 23)
    i32x8 gz = {};
    __builtin_amdgcn_tensor_load_to_lds(g0, g1, g2, g3, gz, 0);
#else
    __builtin_amdgcn_tensor_load_to_lds(g0, g1, g2, g3, 0);
#endif
    __builtin_amdgcn_s_wait_tensorcnt(0);
  }
#else
  // ---- fallback gather: 8 threads/row, same padded LDS layout ----
  {
    int idx = word_indices[ibase + w];
    if (idx < 0) idx = 0;
    const float* src = text + ((long)b * Tc + idx) * Dc;
    float* dst = sRows + w * ROWSTRIDE;
    #pragma unroll
    for (int i = 0; i < 32; ++i) {
      const int off = (i * 8 + p) * 4;   // 8 lanes -> 128B contiguous; 16B-aligned
      const float4 v = *(const float4*)(src + off);
      *(float4*)(dst + colOff(off)) = v; // float4 never straddles a 256 segment
    }
  }
#endif
  __syncthreads();

  // ---- logits: dot(row_w, att_w), 8 lanes/row, stride-8 (conflict-free) ----
  {
    float acc = 0.f;
    const float* r = sRows + w * ROWSTRIDE;
    #pragma unroll 8
    for (int i = 0; i < 128; ++i) {
      const int col  = p + (i << 3);
      const int phys = col + ((i >> 5) << 2);   // == colOff(col) here
      acc += r[phys] * sW[col];
    }
    sPart[tid] = acc;
  }
  __syncthreads();
  if (tid < 16) {
    float acc = 0.f;
    #pragma unroll
    for (int j = 0; j < 8; ++j) acc += sPart[tid * 8 + j];  // deterministic order
    sLog[tid] = acc;
  }
  __syncthreads();

  // ---- masked softmax over W=16 (reference semantics), masks folded in ----
  if (tid == 0) {
    const float bb = att_b[0];
    float ml[16];
    float mx = -3.0e38f;
    #pragma unroll
    for (int i = 0; i < 16; ++i) {
      const int mk = word_mask[ibase + i];
      const float l = (mk != 0) ? (sLog[i] + bb) : -1.0e9f;
      ml[i] = l;
      mx = fmaxf(mx, l);
    }
    float sum = 0.f;
    float e[16];
    #pragma unroll
    for (int i = 0; i < 16; ++i) { e[i] = expf(ml[i] - mx); sum += e[i]; }
    const float inv = 1.0f / sum;
    const float sscale = (word_indices_mask[span] != 0) ? 1.0f : 0.0f;
    #pragma unroll
    for (int i = 0; i < 16; ++i) {
      const int mk = word_mask[ibase + i];
      sProb[i] = ((mk != 0) ? e[i] * inv : 0.0f) * sscale;
    }
  }
  __syncthreads();

  // ---- pooling via V_WMMA_F32_16X16X4_F32 ----
  // C = A(16x4) x B(4x16) over 4 k-chunks; A rows all equal the weights,
  // B[k][n] = E[k][d0+n]. Row 0 of C (c[0], lanes 0-15) is out[d0..d0+15].
  const int wave = tid >> 5;                 // 0..3, 16 d-tiles each
  const int n  = lane & 15;
  const int hk = (lane < 16) ? 0 : 2;        // K-half per documented A layout
  v2f a0, a1, a2, a3;
  a0.x = sProb[0  + hk]; a0.y = sProb[0  + hk + 1];
  a1.x = sProb[4  + hk]; a1.y = sProb[4  + hk + 1];
  a2.x = sProb[8  + hk]; a2.y = sProb[8  + hk + 1];
  a3.x = sProb[12 + hk]; a3.y = sProb[12 + hk + 1];

  float* outSpan = out + (long)span * Dc;
  for (int tile = 0; tile < 16; ++tile) {
    const int d0   = (wave * 16 + tile) * 16;
    const int base = colOff(d0) + n;         // physical column in padded rows
    v8f c = {};
    v2f bv;
    bv.x = sRows[(0  + hk) * ROWSTRIDE + base];
    bv.y = sRows[(1  + hk) * ROWSTRIDE + base];
    c = __builtin_amdgcn_wmma_f32_16x16x4_f32(false, a0, false, bv, (short)0, c, false, false);
    bv.x = sRows[(4  + hk) * ROWSTRIDE + base];
    bv.y = sRows[(5  + hk) * ROWSTRIDE + base];
    c = __builtin_amdgcn_wmma_f32_16x16x4_f32(false, a1, false, bv, (short)0, c, false, false);
    bv.x = sRows[(8  + hk) * ROWSTRIDE + base];
    bv.y = sRows[(9  + hk) * ROWSTRIDE + base];
    c = __builtin_amdgcn_wmma_f32_16x16x4_f32(false, a2, false, bv, (short)0, c, false, false);
    bv.x = sRows[(12 + hk) * ROWSTRIDE + base];
    bv.y = sRows[(13 + hk) * ROWSTRIDE + base];
    c = __builtin_amdgcn_wmma_f32_16x16x4_f32(false, a3, false, bv, (short)0, c, false, false);
    if (lane < 16) outSpan[d0 + n] = c[0];
  }
}

extern "C" void kernel_launch(void* const* d_in, const int* in_sizes, int n_in,
                              void* d_out, int out_size, void* d_ws, size_t ws_size,
                              hipStream_t stream) {
  (void)in_sizes; (void)n_in; (void)out_size; (void)d_ws; (void)ws_size;
  const float* text = (const float*)d_in[0];
  // d_in[1] contextualized_embedding: unused by the reference
  const int* wi  = (const int*)d_in[2];
  const int* wm  = (const int*)d_in[3];
  const int* wim = (const int*)d_in[4];
  const float* aw = (const float*)d_in[5];
  const float* ab = (const float*)d_in[6];
  float* out = (float*)d_out;

  const size_t shmem =
      (size_t)(16 * ROWSTRIDE + 1024 + 128 + 16 + 16) * sizeof(float);
  span_attend_kernel<<<Bc * Sc, 128, shmem, stream>>>(text, wi, wm, wim, aw, ab, out);
}